// DoubleLayeredEncoderCAT_53781580480950
// MI455X (gfx1250) — compile-verified
//
#include <hip/hip_runtime.h>

// ---------------------------------------------------------------------------
// DoubleLayeredEncoderCAT: h = x@W (WMMA f32 16x16x4), edge gather/scale,
// scatter-add (fp32 global atomics into L2-resident d_out with the concat
// remap fused), then bias + PReLU in place.
// ---------------------------------------------------------------------------

typedef __attribute__((ext_vector_type(2))) float v2f;
typedef __attribute__((ext_vector_type(8))) float v8f;

#define N_NODES 100000
#define IN_CH   256
#define HID     64
#define HALF    50000
#define MTILES  (N_NODES / 16)   // 6250, exact

// ---------------------------------------------------------------- zero d_out
__global__ __launch_bounds__(256) void zero_out_kernel(float4* __restrict__ out, int n4) {
    int i = blockIdx.x * blockDim.x + threadIdx.x;
    if (i < n4) out[i] = make_float4(0.f, 0.f, 0.f, 0.f);
}

// ------------------------------------------------------------- h = x @ W
// One wave (32 lanes) computes a 16x64 strip of h using 4 accumulators that
// share a single A fragment per K-step.
//
// W is staged in LDS in a K-quad interleaved layout:
//     lds_wt[(k>>2)*256 + c*4 + (k&3)] = W[k*64 + c]
// so each lane's B fragment (K = khalf, khalf+1 at column c) is ONE contiguous
// 8-byte ds_load_b64, and lanes 0-15 / 16-31 together touch all 64 LDS banks
// exactly once (banks {4c,4c+1} / {4c+2,4c+3}) -> conflict-free.
__global__ __launch_bounds__(256) void gemm_wmma_f32_kernel(const float* __restrict__ x,
                                                            const float* __restrict__ W,
                                                            float* __restrict__ h) {
    __shared__ float lds_wt[IN_CH * HID];   // 64 KB exactly (<= 320 KB WGP LDS)
    for (int i = threadIdx.x; i < IN_CH * HID; i += 256) {
        const int k = i >> 6;
        const int c = i & 63;
        lds_wt[(k >> 2) * 256 + c * 4 + (k & 3)] = W[i];
    }
    __syncthreads();

    const int wid = blockIdx.x * 8 + (threadIdx.x >> 5);   // wave id = M tile
    if (wid >= MTILES) return;

    const int lane  = threadIdx.x & 31;
    const int row0  = wid * 16;
    const int mrow  = lane & 15;              // A: M = lane%16
    const int khalf = (lane >> 4) * 2;        // A/B: lanes 16-31 hold K+2,K+3
    const int col   = lane & 15;              // B/C/D: N = lane%16

    v8f acc0 = {}, acc1 = {}, acc2 = {}, acc3 = {};
    const float* xrow = x + (size_t)(row0 + mrow) * IN_CH;
    // per-lane base into the interleaved W image (provably 8B aligned)
    const float* wt = lds_wt + col * 4 + khalf;

#pragma unroll 4
    for (int k = 0; k < IN_CH; k += 4) {
        // A fragment: 16x4 f32 -> v[0]=x[m,k+khalf], v[1]=x[m,k+khalf+1]
        v2f a = *(const v2f*)(xrow + k + khalf);

        const float* wk = wt + (k >> 2) * 256;
        v2f b0 = *(const v2f*)(wk +   0);   // cols  0..15
        v2f b1 = *(const v2f*)(wk +  64);   // cols 16..31
        v2f b2 = *(const v2f*)(wk + 128);   // cols 32..47
        v2f b3 = *(const v2f*)(wk + 192);   // cols 48..63

        acc0 = __builtin_amdgcn_wmma_f32_16x16x4_f32(false, a, false, b0, (short)0, acc0, false, false);
        acc1 = __builtin_amdgcn_wmma_f32_16x16x4_f32(false, a, false, b1, (short)0, acc1, false, false);
        acc2 = __builtin_amdgcn_wmma_f32_16x16x4_f32(false, a, false, b2, (short)0, acc2, false, false);
        acc3 = __builtin_amdgcn_wmma_f32_16x16x4_f32(false, a, false, b3, (short)0, acc3, false, false);
    }

    // C/D layout: VGPR r -> M = r (lanes 0-15) / r+8 (lanes 16-31), N = lane%16
    const int srow = row0 + (lane >> 4) * 8;
#pragma unroll
    for (int r = 0; r < 8; ++r) {
        float* hr = h + (size_t)(srow + r) * HID + col;
        hr[ 0] = acc0[r];
        hr[16] = acc1[r];
        hr[32] = acc2[r];
        hr[48] = acc3[r];
    }
}

// ----------------------------------------------- scatter: msgs -> segment_sum
// One thread per (edge, channel-pair). Atomic fp32 adds land in d_out, which
// is 25.6 MB and L2-resident (192 MB L2). Concat remap fused into the address.
__global__ __launch_bounds__(256) void scatter_kernel(const int* __restrict__ ei,
                                                      const float* __restrict__ ew,
                                                      const float* __restrict__ h,
                                                      float* __restrict__ out,
                                                      int E) {
    long long idx = (long long)blockIdx.x * blockDim.x + threadIdx.x;
    long long total = (long long)E * (HID / 2);
    if (idx >= total) return;

    const int e = (int)(idx >> 5);          // HID/2 == 32
    const int c = ((int)idx & 31) * 2;

    const int   src = ei[e];
    const int   dst = ei[E + e];
    const float w   = ew[e];

    const float2 hv = *(const float2*)(h + (size_t)src * HID + c);

    // remap: node n -> row n%HALF, col ofs (n/HALF)*64  (the axis-1 concat)
    size_t base = (dst < HALF) ? ((size_t)dst * 128 + c)
                               : ((size_t)(dst - HALF) * 128 + 64 + c);
    atomicAdd(out + base,     w * hv.x);
    atomicAdd(out + base + 1, w * hv.y);
}

// ------------------------------------------------------- bias + PReLU in place
__global__ __launch_bounds__(256) void finalize_kernel(float* __restrict__ out,
                                                       const float* __restrict__ b,
                                                       const float* __restrict__ a,
                                                       int n) {
    int i = blockIdx.x * blockDim.x + threadIdx.x;
    if (i >= n) return;
    const int ch = i & 63;                  // (i%128)%64 == i&63
    const float v = out[i] + b[ch];
    out[i] = (v > 0.f) ? v : a[ch] * v;
}

// ---------------------------------------------------------------------------
extern "C" void kernel_launch(void* const* d_in, const int* in_sizes, int n_in,
                              void* d_out, int out_size, void* d_ws, size_t ws_size,
                              hipStream_t stream) {
    const float* x  = (const float*)d_in[0];
    const int*   ei = (const int*)  d_in[1];   // edge_index, [2, E] flat
    const float* ew = (const float*)d_in[2];   // edge_weight [E]
    const float* W  = (const float*)d_in[3];
    const float* b  = (const float*)d_in[4];
    const float* pa = (const float*)d_in[5];

    float* out = (float*)d_out;                // 50000 x 128 fp32
    float* h   = (float*)d_ws;                 // 100000 x 64 fp32 = 25.6 MB

    const int E = in_sizes[2];                 // edge_weight element count

    // 1) zero accumulator (d_out doubles as segment-sum target)
    const int n4 = out_size / 4;
    zero_out_kernel<<<(n4 + 255) / 256, 256, 0, stream>>>((float4*)out, n4);

    // 2) h = x @ W via fp32 WMMA; 8 waves/block, one 16x64 strip per wave
    gemm_wmma_f32_kernel<<<(MTILES + 7) / 8, 256, 0, stream>>>(x, W, h);

    // 3) gather + scale + scatter-add (fp32 global atomics)
    const long long tot = (long long)E * (HID / 2);
    scatter_kernel<<<(int)((tot + 255) / 256), 256, 0, stream>>>(ei, ew, h, out, E);

    // 4) + b, PReLU, in place
    finalize_kernel<<<(out_size + 255) / 256, 256, 0, stream>>>(out, b, pa, out_size);
}